// NCCLoss_85323820302617
// MI455X (gfx1250) — compile-verified
//
#include <hip/hip_runtime.h>

// ---------------------------------------------------------------------------
// NCC loss, fused single pass for MI455X (gfx1250, wave32).
//
// Box filters run on the matrix pipe: a 9-tap sliding sum over a 16-wide tile
// == 16xK * Kx16 matmul against a constant banded 0/1 matrix, executed as
// chained V_WMMA_F32_16X16X4_F32 accumulations (K in chunks of 4).
// Global->LDS staging uses GLOBAL_LOAD_ASYNC_TO_LDS_B128 (ASYNCcnt) for
// interior tiles; border tiles take a guarded scalar path with zero padding.
// ---------------------------------------------------------------------------

typedef __attribute__((ext_vector_type(2))) float v2f;
typedef __attribute__((ext_vector_type(8))) float v8f;

// Async-to-LDS builtin signature (from clang diagnostic):
//   (int __vector(4) __device__*, int __vector(4) __shared__*, imm, imm)
typedef int v4i_b128 __attribute__((vector_size(4 * sizeof(int))));
typedef v4i_b128 __attribute__((address_space(1))) gb128_t;  // global
typedef v4i_b128 __attribute__((address_space(3))) lb128_t;  // LDS

#define WIDTH    512
#define HEIGHT   512
#define NIMG     32
#define SH_PITCH 148   // 32 x 144 staged region; 16B rows, step 20 mod 64 -> conflict-free
#define PW_PITCH 26    // per-wave scratch pitch; step 26 mod 64 -> conflict-free

static __device__ __forceinline__ v8f wmma_acc(v2f a, v2f b, v8f c) {
  // (neg_a, A, neg_b, B, c_mod, C, reuse_a, reuse_b)
  return __builtin_amdgcn_wmma_f32_16x16x4_f32(false, a, false, b, (short)0, c,
                                               false, false);
}

// H[m0+m][n0+n] = sum_{k=n0+n}^{n0+n+8} X[m0+m][k]   (A = data, B = banded const)
static __device__ __forceinline__ v8f horiz9_tile(const float* X, int xp,
                                                  int m0, int n0, int lane,
                                                  const v2f* band) {
  const int m  = lane & 15;
  const int kh = (lane >> 4) * 2;
  v8f c = {};
#pragma unroll
  for (int t = 0; t < 6; ++t) {
    const int k0 = n0 + 4 * t + kh;
    v2f a;
    a.x = X[(m0 + m) * xp + k0];       // even-aligned pair -> ds_load_b64
    a.y = X[(m0 + m) * xp + k0 + 1];
    c = wmma_acc(a, band[t], c);
  }
  return c;
}

// S[m0+m][n0+n] = sum_{k=m0+m}^{m0+m+8} H[k][n0+n]   (A = banded const, B = data)
static __device__ __forceinline__ v8f vert9_tile(const float* Hb, int hp,
                                                 int m0, int n0, int lane,
                                                 const v2f* band) {
  const int mn = lane & 15;
  const int kh = (lane >> 4) * 2;
  v8f c = {};
#pragma unroll
  for (int t = 0; t < 6; ++t) {
    const int k0 = m0 + 4 * t + kh;
    v2f b;
    b.x = Hb[k0 * hp + n0 + mn];
    b.y = Hb[(k0 + 1) * hp + n0 + mn];
    c = wmma_acc(band[t], b, c);
  }
  return c;
}

// D-matrix layout: VGPR v -> row v (lanes 0-15) / row v+8 (lanes 16-31), col lane&15.
static __device__ __forceinline__ void store16x16(float* dst, int dp, int r0,
                                                  int c0, v8f c, int lane) {
  const int n  = lane & 15;
  const int rh = (lane >> 4) * 8;
#pragma unroll
  for (int v = 0; v < 8; ++v)
    dst[(r0 + rh + v) * dp + c0 + n] = c[v];
}

__global__ void ncc_fused_kernel(const float* __restrict__ I,
                                 const float* __restrict__ J,
                                 float* __restrict__ acc) {
  extern __shared__ float smem[];
  const int tid  = threadIdx.x;
  const int lane = tid & 31;
  const int w    = tid >> 5;  // wave id 0..7, one 16x16 output tile each

  // 4096 blocks: 32 images x 32 tile-rows x 4 block-cols (block = 8 tiles wide)
  const int b   = blockIdx.x;
  const int n   = b >> 7;
  const int rem = b & 127;
  const int tr  = rem >> 2;
  const int bc  = rem & 3;
  const int r0  = tr * 16;
  const int c0b = bc * 128;

  float* XI    = smem;                   // 32 x SH_PITCH staged region (I)
  float* XJ    = XI + 32 * SH_PITCH;     // same for J
  float* wbase = XJ + 32 * SH_PITCH;     // per-wave scratch, 3328 floats each
  float* HB = wbase + w * 3328;          // 32 x 26 horizontal-pass scratch
  float* SI = HB + 32 * PW_PITCH;        // 24 x 26 box-sum(I); later product buf
  float* SJ = SI + 24 * PW_PITCH;        // 24 x 26 box-sum(J)
  float* DI = SJ + 24 * PW_PITCH;        // 24 x 26 dI
  float* DJ = DI + 24 * PW_PITCH;        // 24 x 26 dJ
  float* red = wbase + 8 * 3328;         // 8-float block-reduce slot

  // Banded 0/1 fragments: value = [ (4t + kh - (lane&15)) in [0,8] ].
  // Identical for horizontal-B and vertical-A roles, for every tile origin.
  v2f band[6];
  {
    const int mn = lane & 15;
    const int kh = (lane >> 4) * 2;
#pragma unroll
    for (int t = 0; t < 6; ++t) {
      const int d = 4 * t + kh - mn;
      band[t].x = (d >= 0 && d <= 8) ? 1.0f : 0.0f;
      band[t].y = (d + 1 >= 0 && d + 1 <= 8) ? 1.0f : 0.0f;
    }
  }

  // --- stage 0: stage the 32x144 region (with 8-halo) into LDS ---
  const size_t ibase = (size_t)n * (WIDTH * HEIGHT);
  const bool interior = (tr >= 1) && (tr <= 30) && (bc >= 1) && (bc <= 2);
  if (interior) {
    // All addresses in-bounds and 16B-aligned: async b128 copies straight to
    // LDS (no VGPR round-trip), tracked on ASYNCcnt.
    const float* Ib = I + ibase + (size_t)(r0 - 8) * WIDTH + (c0b - 8);
    const float* Jb = J + ibase + (size_t)(r0 - 8) * WIDTH + (c0b - 8);
    for (int idx = tid; idx < 32 * 36; idx += 256) {  // 36 float4 per row
      const int rr = idx / 36, c4 = idx % 36;
      float*       lI = &XI[rr * SH_PITCH + c4 * 4];
      float*       lJ = &XJ[rr * SH_PITCH + c4 * 4];
      const float* gI = Ib + rr * WIDTH + c4 * 4;
      const float* gJ = Jb + rr * WIDTH + c4 * 4;
#if __has_builtin(__builtin_amdgcn_global_load_async_to_lds_b128)
      __builtin_amdgcn_global_load_async_to_lds_b128(
          (gb128_t*)gI, (lb128_t*)lI, 0, 0);
      __builtin_amdgcn_global_load_async_to_lds_b128(
          (gb128_t*)gJ, (lb128_t*)lJ, 0, 0);
#else
      *(float4*)lI = *(const float4*)gI;
      *(float4*)lJ = *(const float4*)gJ;
#endif
    }
#if __has_builtin(__builtin_amdgcn_global_load_async_to_lds_b128)
#if __has_builtin(__builtin_amdgcn_s_wait_asynccnt)
    __builtin_amdgcn_s_wait_asynccnt(0);
#else
    asm volatile("s_wait_asynccnt 0x0" ::: "memory");
#endif
#endif
  } else {
    // Border blocks: guarded scalar path with zero padding.
    for (int idx = tid; idx < 32 * 144; idx += 256) {
      const int rr = idx / 144, cc = idx % 144;
      const int gr = r0 - 8 + rr;
      const int gc = c0b - 8 + cc;
      const bool ok = (gr >= 0) && (gr < HEIGHT) && (gc >= 0) && (gc < WIDTH);
      const size_t g = ibase + (size_t)gr * WIDTH + (size_t)gc;
      XI[rr * SH_PITCH + cc] = ok ? I[g] : 0.0f;
      XJ[rr * SH_PITCH + cc] = ok ? J[g] : 0.0f;
    }
  }
  __syncthreads();

  // Per-wave 32x32 sub-window (16-col offset per wave; rows shared block-wide).
  const float* XIw = XI + 16 * w;
  const float* XJw = XJ + 16 * w;

  // --- stage 1: box sums of I and J on a 24x24 block (tile + 4 halo) ---
#pragma unroll
  for (int m0 = 0; m0 < 32; m0 += 16)
#pragma unroll
    for (int n0 = 0; n0 <= 8; n0 += 8)
      store16x16(HB, PW_PITCH, m0, n0,
                 horiz9_tile(XIw, SH_PITCH, m0, n0, lane, band), lane);
#pragma unroll
  for (int m0 = 0; m0 <= 8; m0 += 8)
#pragma unroll
    for (int n0 = 0; n0 <= 8; n0 += 8)
      store16x16(SI, PW_PITCH, m0, n0,
                 vert9_tile(HB, PW_PITCH, m0, n0, lane, band), lane);

#pragma unroll
  for (int m0 = 0; m0 < 32; m0 += 16)
#pragma unroll
    for (int n0 = 0; n0 <= 8; n0 += 8)
      store16x16(HB, PW_PITCH, m0, n0,
                 horiz9_tile(XJw, SH_PITCH, m0, n0, lane, band), lane);
#pragma unroll
  for (int m0 = 0; m0 <= 8; m0 += 8)
#pragma unroll
    for (int n0 = 0; n0 <= 8; n0 += 8)
      store16x16(SJ, PW_PITCH, m0, n0,
                 vert9_tile(HB, PW_PITCH, m0, n0, lane, band), lane);

  // --- stage 2: dI = I - mean(I), dJ = J - mean(J) on the 24x24 block ---
  for (int idx = lane; idx < 576; idx += 32) {
    const int i = idx / 24, j = idx % 24;
    const float mI = SI[i * PW_PITCH + j] * (1.0f / 81.0f);
    const float mJ = SJ[i * PW_PITCH + j] * (1.0f / 81.0f);
    DI[i * PW_PITCH + j] = XIw[(i + 4) * SH_PITCH + (j + 4)] - mI;
    DJ[i * PW_PITCH + j] = XJw[(i + 4) * SH_PITCH + (j + 4)] - mJ;
  }

  // --- stage 3: box sums of dI*dI, dJ*dJ, dI*dJ -> final 16x16 in registers ---
  auto boxprod = [&](const float* Pa, const float* Pb) -> v8f {
    for (int idx = lane; idx < 576; idx += 32) {
      const int i = idx / 24, j = idx % 24;
      SI[i * PW_PITCH + j] = Pa[i * PW_PITCH + j] * Pb[i * PW_PITCH + j];
    }
#pragma unroll
    for (int m0 = 0; m0 <= 8; m0 += 8)
      store16x16(HB, PW_PITCH, m0, 0,
                 horiz9_tile(SI, PW_PITCH, m0, 0, lane, band), lane);
    return vert9_tile(HB, PW_PITCH, 0, 0, lane, band);
  };
  const v8f vIs = boxprod(DI, DI);
  const v8f vJs = boxprod(DJ, DJ);
  const v8f cvs = boxprod(DI, DJ);

  // --- stage 4: ncc per pixel (values live in the D-matrix regs), reduce ---
  float s = 0.0f;
#pragma unroll
  for (int v = 0; v < 8; ++v) {
    const float varI = vIs[v] * (1.0f / 81.0f);
    const float varJ = vJs[v] * (1.0f / 81.0f);
    const float cov  = cvs[v] * (1.0f / 81.0f);
    s += (cov * cov) / (varI * varJ + 1e-5f);
  }
#pragma unroll
  for (int off = 16; off > 0; off >>= 1) s += __shfl_xor(s, off, 32);
  if (lane == 0) red[w] = s;
  __syncthreads();
  if (tid == 0) {
    float t = 0.0f;
#pragma unroll
    for (int k = 0; k < 8; ++k) t += red[k];
    atomicAdd(acc, t);
  }
}

__global__ void ncc_finalize(const float* __restrict__ acc,
                             float* __restrict__ out) {
  out[0] = -acc[0] * (1.0f / (float)(NIMG * WIDTH * HEIGHT));
}

extern "C" void kernel_launch(void* const* d_in, const int* in_sizes, int n_in,
                              void* d_out, int out_size, void* d_ws,
                              size_t ws_size, hipStream_t stream) {
  (void)in_sizes; (void)n_in; (void)out_size; (void)ws_size;
  const float* I = (const float*)d_in[0];
  const float* J = (const float*)d_in[1];
  float* acc = (float*)d_ws;

  (void)hipMemsetAsync(acc, 0, sizeof(float), stream);  // capture-safe

  const size_t shbytes =
      (size_t)(2 * 32 * SH_PITCH + 8 * 3328 + 8) * sizeof(float);
  ncc_fused_kernel<<<4096, 256, shbytes, stream>>>(I, J, acc);
  ncc_finalize<<<1, 1, 0, stream>>>(acc, (float*)d_out);
}